// DySnakeConv_22385369547426
// MI455X (gfx1250) — compile-verified
//
#include <hip/hip_runtime.h>
#include <hip/hip_bf16.h>

#define BATCH 8
#define CIN   64
#define COUTM 128
#define HD    128
#define WD    128
#define KK    9
#define KDIM  (CIN*KK)      // 576
#define KSTEPS (KDIM/4)     // 144
#define EPSV  1e-5f

typedef float v2f __attribute__((ext_vector_type(2)));
typedef float v8f __attribute__((ext_vector_type(8)));

static __device__ __forceinline__ v8f wmma4(v2f a, v2f b, v8f c) {
  // D = A(16x4 f32) * B(4x16 f32) + C(16x16 f32), fp32 on the matrix pipe
  return __builtin_amdgcn_wmma_f32_16x16x4_f32(false, a, false, b, (short)0, c, false, false);
}

static __device__ __forceinline__ float fast_sigmoid(float v) {
  return __builtin_amdgcn_rcpf(1.f + __expf(-v));   // v_rcp_f32 + v_exp_f32
}

// ---------------------------------------------------------------------------
// Repack conv weights (Msrc x 576, K = ci*9 + (ky*3+kx)) into A-fragment order:
//   dst[(ks*Mtiles + mt)*32 + lane] = float2{ W[m][kb], W[m][kb+1] }
//   with m = mt*16 + (lane&15), kb = 4*ks + 2*(lane>=16)
// matching the ISA 32-bit A 16x4 layout (vgpr0: K=0/K=2, vgpr1: K=1/K=3).
// ---------------------------------------------------------------------------
__global__ __launch_bounds__(256) void repack_weights(
    const float* __restrict__ src, float* __restrict__ dst, int Msrc, int Mtiles) {
  int idx = blockIdx.x * blockDim.x + threadIdx.x;
  int total = KSTEPS * Mtiles * 32;
  if (idx >= total) return;
  int lane = idx & 31;
  int mt   = (idx >> 5) % Mtiles;
  int ks   = idx / (32 * Mtiles);
  int m    = mt * 16 + (lane & 15);
  int kb   = ks * 4 + ((lane >> 4) << 1);
  float2 v = make_float2(0.f, 0.f);
  if (m < Msrc) { v.x = src[m * KDIM + kb]; v.y = src[m * KDIM + kb + 1]; }
  reinterpret_cast<float2*>(dst)[idx] = v;
}

// ---------------------------------------------------------------------------
// Training-mode BN statistics per channel, packed with gamma/beta:
// bnp[{mean, rstd, gamma, beta} * CIN]
// ---------------------------------------------------------------------------
__global__ __launch_bounds__(256) void bn_stats(
    const float* __restrict__ t, const float* __restrict__ gamma,
    const float* __restrict__ beta, float* __restrict__ bnp) {
  __shared__ float ss[256], ss2[256];
  const int c = blockIdx.x;
  const int tid = threadIdx.x;
  const int NPC = BATCH * HD * WD;   // 131072
  float s = 0.f, s2 = 0.f;
  for (int i = tid; i < NPC; i += 256) {
    int b = i >> 14;
    int p = i & 16383;
    float v = t[(size_t)(b * CIN + c) * (HD * WD) + p];
    s += v; s2 += v * v;
  }
  ss[tid] = s; ss2[tid] = s2;
  __syncthreads();
  for (int o = 128; o > 0; o >>= 1) {
    if (tid < o) { ss[tid] += ss[tid + o]; ss2[tid] += ss2[tid + o]; }
    __syncthreads();
  }
  if (tid == 0) {
    float inv = 1.f / (float)NPC;
    float m = ss[0] * inv;
    float var = ss2[0] * inv - m * m;
    bnp[c]           = m;
    bnp[CIN + c]     = rsqrtf(var + EPSV);
    bnp[2 * CIN + c] = gamma[c];
    bnp[3 * CIN + c] = beta[c];
  }
}

// ---------------------------------------------------------------------------
// Generic 3x3 conv, Cin=64, implicit GEMM M=Cout(<=64) x K=576 x Ntile=32.
// Stage 1: global -> LDS halo patch (BN+SiLU applied in-bounds; halo = 0).
// Stage 2: patch -> LDS B-fragment panel (all tap/channel math done once).
// Stage 3: K-loop = b64(A) + b64(B) + wmma, dual accumulators for ILP.
// Dynamic LDS: patch 27,648 B + panel 73,728 B.
// ---------------------------------------------------------------------------
__global__ __launch_bounds__(256) void conv3x3_wmma(
    const float* __restrict__ in, const float* __restrict__ wA,
    const float* __restrict__ bias, float* __restrict__ out,
    const float* __restrict__ bnp, int Mtiles, int Msrc, int act) {
  extern __shared__ float lds[];
  float* patch = lds;                         // [ci][ky][36] cols x0-1..x0+32
  float* panel = lds + CIN * 3 * 36;          // fragment order: [(ks*2+nsub)*32+lane] float2
  const int tid  = threadIdx.x;
  const int lane = tid & 31;
  const int wv   = tid >> 5;
  const int x0   = blockIdx.x * 32;
  const int y    = blockIdx.y;
  const int b    = blockIdx.z;

  // Stage 1: halo patch with optional BN+SiLU (padding stays 0, matching ref)
  for (int e = tid; e < CIN * 3 * 34; e += 256) {
    int c  = e % 34;
    int r  = (e / 34) % 3;
    int ci = e / (34 * 3);
    int gx = x0 - 1 + c;
    int gy = y - 1 + r;
    float v = 0.f;
    if (gx >= 0 && gx < WD && gy >= 0 && gy < HD) {
      v = in[((size_t)(b * CIN + ci) * HD + gy) * WD + gx];
      if (bnp) {
        v = (v - bnp[ci]) * bnp[CIN + ci] * bnp[2 * CIN + ci] + bnp[3 * CIN + ci];
        v = v * fast_sigmoid(v);   // SiLU
      }
    }
    patch[(ci * 3 + r) * 36 + c] = v;
  }
  __syncthreads();

  // Stage 2: build B-fragment panel (one-time tap/channel decomposition)
  for (int i = tid; i < KSTEPS * 2 * 32; i += 256) {
    int l    = i & 31;
    int nsub = (i >> 5) & 1;
    int ks   = i >> 6;
    int nl   = l & 15;
    int K0   = ks * 4 + ((l >> 4) << 1);
    float2 v;
    {
      int K = K0;
      int ci = K / 9, t = K - 9 * ci, ky = t / 3, kx = t - 3 * ky;
      v.x = patch[(ci * 3 + ky) * 36 + nsub * 16 + nl + kx];
    }
    {
      int K = K0 + 1;
      int ci = K / 9, t = K - 9 * ci, ky = t / 3, kx = t - 3 * ky;
      v.y = patch[(ci * 3 + ky) * 36 + nsub * 16 + nl + kx];
    }
    reinterpret_cast<float2*>(panel)[i] = v;
  }
  __syncthreads();

  // Stage 3: WMMA K-loop, linear addressing, two independent accumulators
  const v2f* wA2 = reinterpret_cast<const v2f*>(wA);
  const v2f* bp  = reinterpret_cast<const v2f*>(panel);
  const bool full = (Msrc == Mtiles * 16);
  for (int tile = wv; tile < Mtiles * 2; tile += 8) {
    int mt   = tile >> 1;
    int nsub = tile & 1;
    v8f acc0 = {}, acc1 = {};
    #pragma unroll 4
    for (int ks = 0; ks < KSTEPS; ks += 2) {
      v2f a0 = wA2[( ks      * Mtiles + mt) * 32 + lane];
      v2f b0 = bp [( ks * 2      + nsub) * 32 + lane];
      v2f a1 = wA2[((ks + 1) * Mtiles + mt) * 32 + lane];
      v2f b1 = bp [((ks + 1) * 2 + nsub) * 32 + lane];
      acc0 = wmma4(a0, b0, acc0);
      acc1 = wmma4(a1, b1, acc1);
    }
    v8f acc = acc0 + acc1;
    int n = x0 + nsub * 16 + (lane & 15);
    int mbase = mt * 16 + ((lane >> 4) << 3);   // C: vgpr v -> M=v (lo lanes) / v+8 (hi)
    if (full) {
      #pragma unroll
      for (int v = 0; v < 8; ++v) {
        int m = mbase + v;
        float r = acc[v] + bias[m];
        if (act == 1) r = fast_sigmoid(r);
        out[((size_t)(b * Msrc + m) * HD + y) * WD + n] = r;
      }
    } else {
      #pragma unroll
      for (int v = 0; v < 8; ++v) {
        int m = mbase + v;
        if (m < Msrc) {
          float r = acc[v] + bias[m];
          if (act == 1) r = fast_sigmoid(r);
          out[((size_t)(b * Msrc + m) * HD + y) * WD + n] = r;
        }
      }
    }
  }
}

// ---------------------------------------------------------------------------
// Deformable sampling + final contraction (Cout=128, K=576, Ntile=16).
// Phase 1: gather modulated bilinear samples straight into B-fragment slots.
// Phase 2: GEMM 128x576x16, one 16x16 C tile per wave, dual accumulators.
// ---------------------------------------------------------------------------
__global__ __launch_bounds__(256) void deform_gemm(
    const float* __restrict__ x, const float* __restrict__ offs,
    const float* __restrict__ modu, const float* __restrict__ wA,
    const float* __restrict__ bias, float* __restrict__ out) {
  __shared__ float smp[KSTEPS * 32 * 2];   // 36,864 B, fragment-ordered
  const int tid  = threadIdx.x;
  const int lane = tid & 31;
  const int wv   = tid >> 5;
  const int x0   = blockIdx.x * 16;
  const int y    = blockIdx.y;
  const int b    = blockIdx.z;
  const int HW   = HD * WD;

  // Phase 1: 9 taps x 16 pixels x 64 channels = 9216 gathers
  for (int it = tid; it < 9 * 16 * CIN; it += 256) {
    int ci   = it & 63;
    int pair = it >> 6;      // 0..143
    int t    = pair >> 4;    // tap 0..8
    int n    = pair & 15;    // pixel in tile
    int xx   = x0 + n;
    int ky = t / 3, kx = t - ky * 3;
    float offx = offs[((size_t)(b * 18 + 2 * t    ) * HD + y) * WD + xx];
    float offy = offs[((size_t)(b * 18 + 2 * t + 1) * HD + y) * WD + xx];
    float mo   = modu[((size_t)(b * 9 + t) * HD + y) * WD + xx];
    float sy = (float)(y + ky - 1) + offy;
    float sx = (float)(xx + kx - 1) + offx;
    float fy = floorf(sy), fx = floorf(sx);
    float wy1 = sy - fy, wx1 = sx - fx;
    int y0 = (int)fy, x0i = (int)fx;
    int y1 = y0 + 1, x1 = x0i + 1;
    bool vy0 = (y0 >= 0) & (y0 < HD), vy1 = (y1 >= 0) & (y1 < HD);
    bool vx0 = (x0i >= 0) & (x0i < WD), vx1 = (x1 >= 0) & (x1 < WD);
    int cy0 = min(max(y0, 0), HD - 1), cy1 = min(max(y1, 0), HD - 1);
    int cx0 = min(max(x0i, 0), WD - 1), cx1 = min(max(x1, 0), WD - 1);
    float w00 = (vy0 && vx0) ? (1.f - wy1) * (1.f - wx1) : 0.f;
    float w01 = (vy0 && vx1) ? (1.f - wy1) * wx1 : 0.f;
    float w10 = (vy1 && vx0) ? wy1 * (1.f - wx1) : 0.f;
    float w11 = (vy1 && vx1) ? wy1 * wx1 : 0.f;
    const float* img = x + (size_t)(b * CIN + ci) * HW;
    float val = w00 * img[cy0 * WD + cx0] + w01 * img[cy0 * WD + cx1]
              + w10 * img[cy1 * WD + cx0] + w11 * img[cy1 * WD + cx1];
    // scatter into B-fragment slot: K = ci*9 + t
    int K  = ci * 9 + t;
    int ks = K >> 2;
    int fl = ((K & 2) << 3) + n;   // lane within fragment
    int j  = K & 1;
    smp[(ks * 32 + fl) * 2 + j] = val * mo;
  }
  __syncthreads();

  // Phase 2: per-wave 16x16 C tile, 144 WMMA issues, dual accumulators
  const v2f* wA2 = reinterpret_cast<const v2f*>(wA);
  const v2f* sp  = reinterpret_cast<const v2f*>(smp);
  const int mt = wv;                   // 8 waves cover M=128
  v8f acc0 = {}, acc1 = {};
  #pragma unroll 4
  for (int ks = 0; ks < KSTEPS; ks += 2) {
    v2f a0 = wA2[( ks      * 8 + mt) * 32 + lane];
    v2f b0 = sp [  ks           * 32 + lane];
    v2f a1 = wA2[((ks + 1) * 8 + mt) * 32 + lane];
    v2f b1 = sp [ (ks + 1)      * 32 + lane];
    acc0 = wmma4(a0, b0, acc0);
    acc1 = wmma4(a1, b1, acc1);
  }
  v8f acc = acc0 + acc1;
  int n = x0 + (lane & 15);
  int mbase = mt * 16 + ((lane >> 4) << 3);
  #pragma unroll
  for (int v = 0; v < 8; ++v) {
    int m = mbase + v;
    out[((size_t)(b * COUTM + m) * HD + y) * WD + n] = acc[v] + bias[m];
  }
}

// ---------------------------------------------------------------------------
extern "C" void kernel_launch(void* const* d_in, const int* in_sizes, int n_in,
                              void* d_out, int out_size, void* d_ws, size_t ws_size,
                              hipStream_t stream) {
  const float* x      = (const float*)d_in[0];
  const float* weight = (const float*)d_in[1];
  const float* bias   = (const float*)d_in[2];
  const float* off_w1 = (const float*)d_in[3];
  const float* off_b1 = (const float*)d_in[4];
  const float* off_g  = (const float*)d_in[5];
  const float* off_bt = (const float*)d_in[6];
  const float* off_w2 = (const float*)d_in[7];
  const float* off_b2 = (const float*)d_in[8];
  const float* mod_w1 = (const float*)d_in[9];
  const float* mod_b1 = (const float*)d_in[10];
  const float* mod_g  = (const float*)d_in[11];
  const float* mod_bt = (const float*)d_in[12];
  const float* mod_w2 = (const float*)d_in[13];
  const float* mod_b2 = (const float*)d_in[14];

  float* ws = (float*)d_ws;
  size_t pos = 0;
  auto alloc = [&](size_t n) { float* p = ws + pos; pos += n; return p; };
  float* t1_off  = alloc((size_t)BATCH * CIN * HD * WD);
  float* t1_mod  = alloc((size_t)BATCH * CIN * HD * WD);
  float* offbuf  = alloc((size_t)BATCH * 18 * HD * WD);
  float* modbuf  = alloc((size_t)BATCH * 9 * HD * WD);
  float* wA_off1 = alloc((size_t)KSTEPS * 4 * 32 * 2);
  float* wA_mod1 = alloc((size_t)KSTEPS * 4 * 32 * 2);
  float* wA_off2 = alloc((size_t)KSTEPS * 2 * 32 * 2);
  float* wA_mod2 = alloc((size_t)KSTEPS * 1 * 32 * 2);
  float* wA_main = alloc((size_t)KSTEPS * 8 * 32 * 2);
  float* bnp_off = alloc(4 * CIN);
  float* bnp_mod = alloc(4 * CIN);

  auto rp = [&](const float* src, float* dst, int Msrc, int Mtiles) {
    int total = KSTEPS * Mtiles * 32;
    repack_weights<<<(total + 255) / 256, 256, 0, stream>>>(src, dst, Msrc, Mtiles);
  };
  rp(off_w1, wA_off1, 64, 4);
  rp(mod_w1, wA_mod1, 64, 4);
  rp(off_w2, wA_off2, 18, 2);
  rp(mod_w2, wA_mod2, 9, 1);
  rp(weight, wA_main, 128, 8);

  const size_t conv_lds = (size_t)(CIN * 3 * 36 + KSTEPS * 2 * 32 * 2) * sizeof(float);
  dim3 cgrid(WD / 32, HD, BATCH);
  conv3x3_wmma<<<cgrid, 256, conv_lds, stream>>>(x, wA_off1, off_b1, t1_off, nullptr, 4, 64, 0);
  conv3x3_wmma<<<cgrid, 256, conv_lds, stream>>>(x, wA_mod1, mod_b1, t1_mod, nullptr, 4, 64, 0);
  bn_stats<<<CIN, 256, 0, stream>>>(t1_off, off_g, off_bt, bnp_off);
  bn_stats<<<CIN, 256, 0, stream>>>(t1_mod, mod_g, mod_bt, bnp_mod);
  conv3x3_wmma<<<cgrid, 256, conv_lds, stream>>>(t1_off, wA_off2, off_b2, offbuf, bnp_off, 2, 18, 0);
  conv3x3_wmma<<<cgrid, 256, conv_lds, stream>>>(t1_mod, wA_mod2, mod_b2, modbuf, bnp_mod, 1, 9, 1);

  dim3 dgrid(WD / 16, HD, BATCH);
  deform_gemm<<<dgrid, 256, 0, stream>>>(x, offbuf, modbuf, wA_main, bias, (float*)d_out);
}